// SelfAttention_49624052138252
// MI455X (gfx1250) — compile-verified
//
#include <hip/hip_runtime.h>
#include <hip/hip_bf16.h>
#include <math.h>

// ---------------------------------------------------------------------------
// fp32 WMMA GEMM for gfx1250 (V_WMMA_F32_16X16X4_F32), wave32.
//   C[m,n] = sum_k A[m,k] * B(k,n) (+ bias[n])
//   TRANS_B=false: B(k,n) = B[k*ldb + n]   (row-major KxN)
//   TRANS_B=true : B(k,n) = B[n*ldb + k]   (row-major NxK, read transposed)
// Block tile 128x128, K-tile 16, double-buffered LDS, 8 waves (4x2),
// wave tile 32x64 -> 2x4 fragments of 16x16.
//
// B is stored in LDS K-paired: Bs[kpair][n][2] so that each WMMA B-fragment
// (B(k,n), B(k+1,n)) is one aligned 8-byte LDS load directly into an even
// VGPR pair -> no v_mov marshaling in the inner loop.
// ---------------------------------------------------------------------------

typedef float v2f __attribute__((ext_vector_type(2)));
typedef float v8f __attribute__((ext_vector_type(8)));

#define BM 128
#define BN 128
#define BK 16
#define APAD 4   // A row = 20 floats (80B, 16B aligned; 20*r distinct banks mod 64)
#define BPADN 16 // B pair-row = (128+16)*2 dwords = 288 ; 288 mod 64 = 32 ->
                 // lanes 0-15 (pair p) and 16-31 (pair p+1) hit disjoint banks

template <bool TRANS_B, bool HAS_BIAS>
__global__ __launch_bounds__(256) void wmma_gemm_f32(
    const float* __restrict__ A, const float* __restrict__ B,
    const float* __restrict__ bias, float* __restrict__ C,
    int K, int lda, int ldb, int ldc,
    long long strideA, long long strideB, long long strideC) {
  __shared__ float As[2][BM][BK + APAD];
  __shared__ float Bs[2][BK / 2][BN + BPADN][2];

  const int tid  = threadIdx.x;
  const int lane = tid & 31;
  const int wave = tid >> 5;
  const int waveM = wave >> 1;  // 0..3
  const int waveN = wave & 1;   // 0..1

  const int mTile = blockIdx.x * BM;
  const int nTile = blockIdx.y * BN;

  const float* Ab = A + (long long)blockIdx.z * strideA;
  const float* Bb = B + (long long)blockIdx.z * strideB;
  float*       Cb = C + (long long)blockIdx.z * strideC;

  v8f acc[2][4];
#pragma unroll
  for (int i = 0; i < 2; ++i)
#pragma unroll
    for (int j = 0; j < 4; ++j)
      acc[i][j] = (v8f){0.f, 0.f, 0.f, 0.f, 0.f, 0.f, 0.f, 0.f};

  // ---- tile loaders: 256 threads, float4 global loads -----------------
  auto loadA = [&](int buf, int kt) {
    const int quad = tid & 3;        // k-chunk of 4 floats
    const int row0 = tid >> 2;       // 0..63
    const float* src = Ab + (long long)mTile * lda + kt * BK;
#pragma unroll
    for (int i = 0; i < 2; ++i) {
      const int r = row0 + i * 64;
      float4 v = *reinterpret_cast<const float4*>(src + (long long)r * lda + quad * 4);
      *reinterpret_cast<float4*>(&As[buf][r][quad * 4]) = v;
    }
  };

  auto loadB = [&](int buf, int kt) {
    if (!TRANS_B) {
      const int quadN = tid & 31;    // n-chunk of 4
      const int k0    = tid >> 5;    // 0..7
#pragma unroll
      for (int i = 0; i < 2; ++i) {
        const int k = k0 + i * 8;
        float4 v = *reinterpret_cast<const float4*>(
            Bb + (long long)(kt * BK + k) * ldb + nTile + quadN * 4);
        Bs[buf][k >> 1][quadN * 4 + 0][k & 1] = v.x;
        Bs[buf][k >> 1][quadN * 4 + 1][k & 1] = v.y;
        Bs[buf][k >> 1][quadN * 4 + 2][k & 1] = v.z;
        Bs[buf][k >> 1][quadN * 4 + 3][k & 1] = v.w;
      }
    } else {
      const int quadK = tid & 3;     // k-chunk of 4 (covers K pairs 2q, 2q+1)
      const int n0    = tid >> 2;    // 0..63
#pragma unroll
      for (int i = 0; i < 2; ++i) {
        const int n = n0 + i * 64;
        float4 v = *reinterpret_cast<const float4*>(
            Bb + (long long)(nTile + n) * ldb + kt * BK + quadK * 4);
        *reinterpret_cast<float2*>(&Bs[buf][quadK * 2 + 0][n][0]) =
            make_float2(v.x, v.y);
        *reinterpret_cast<float2*>(&Bs[buf][quadK * 2 + 1][n][0]) =
            make_float2(v.z, v.w);
      }
    }
  };

  // ---- WMMA compute over one K-tile -----------------------------------
  // ISA fragment layouts (f32 16x16x4):
  //   A 16x4 : lane<16 -> row=lane,    {v0,v1}={K0,K1}; lane>=16 -> {K2,K3}
  //   B 4x16 : lane<16 -> col=lane,    {v0,v1}={K0,K1}; lane>=16 -> {K2,K3}
  //   C 16x16: col=lane&15, VGPR i -> row = i + (lane>=16 ? 8 : 0)
  const int fragRow = waveM * 32 + (lane & 15);
  const int fragCol = waveN * 64 + (lane & 15);
  const int kShift  = (lane >> 4) * 2;    // K offset for this half-wave
  const int pShift  = (lane >> 4);        // K-pair offset

  auto compute = [&](int buf) {
#pragma unroll
    for (int kk = 0; kk < BK; kk += 4) {
      const int kb = kk + kShift;
      const int kp = (kk >> 1) + pShift;
      v2f a[2], b[4];
#pragma unroll
      for (int i = 0; i < 2; ++i)
        a[i] = *reinterpret_cast<const v2f*>(&As[buf][fragRow + i * 16][kb]);
#pragma unroll
      for (int j = 0; j < 4; ++j)
        b[j] = *reinterpret_cast<const v2f*>(&Bs[buf][kp][fragCol + j * 16][0]);
#pragma unroll
      for (int i = 0; i < 2; ++i)
#pragma unroll
        for (int j = 0; j < 4; ++j)
          acc[i][j] = __builtin_amdgcn_wmma_f32_16x16x4_f32(
              false, a[i], false, b[j], (short)0, acc[i][j], false, false);
    }
  };

  // ---- main loop: double-buffered -------------------------------------
  const int numK = K / BK;
  loadA(0, 0);
  loadB(0, 0);
  __syncthreads();
  for (int kt = 0; kt < numK; ++kt) {
    const int buf = kt & 1;
    if (kt + 1 < numK) {
      loadA(buf ^ 1, kt + 1);
      loadB(buf ^ 1, kt + 1);
    }
    compute(buf);
    __syncthreads();
  }

  // ---- store (+ optional bias) ----------------------------------------
#pragma unroll
  for (int i = 0; i < 2; ++i) {
#pragma unroll
    for (int j = 0; j < 4; ++j) {
      const int n  = nTile + waveN * 64 + j * 16 + (lane & 15);
      const int m0 = mTile + waveM * 32 + i * 16 + ((lane >> 4) << 3);
      float bv = 0.0f;
      if (HAS_BIAS) bv = bias[n];
#pragma unroll
      for (int r = 0; r < 8; ++r)
        Cb[(long long)(m0 + r) * ldc + n] = acc[i][j][r] + bv;
    }
  }
}

// ---------------------------------------------------------------------------
// Row softmax over rows of length 2048 (one 256-thread block per row).
// ---------------------------------------------------------------------------
__global__ __launch_bounds__(256) void softmax_rows2048(float* __restrict__ S) {
  float* row = S + (long long)blockIdx.x * 2048;
  const int t = threadIdx.x;
  const int lane = t & 31;
  const int wave = t >> 5;

  float v[8];
  float mx = -INFINITY;
#pragma unroll
  for (int i = 0; i < 8; ++i) {
    v[i] = row[t + i * 256];
    mx = fmaxf(mx, v[i]);
  }
#pragma unroll
  for (int off = 16; off > 0; off >>= 1) mx = fmaxf(mx, __shfl_xor(mx, off, 32));

  __shared__ float redmax[8];
  __shared__ float redsum[8];
  if (lane == 0) redmax[wave] = mx;
  __syncthreads();
  mx = redmax[0];
#pragma unroll
  for (int w = 1; w < 8; ++w) mx = fmaxf(mx, redmax[w]);

  float s = 0.0f;
#pragma unroll
  for (int i = 0; i < 8; ++i) {
    v[i] = __expf(v[i] - mx);
    s += v[i];
  }
#pragma unroll
  for (int off = 16; off > 0; off >>= 1) s += __shfl_xor(s, off, 32);
  if (lane == 0) redsum[wave] = s;
  __syncthreads();
  s = 0.0f;
#pragma unroll
  for (int w = 0; w < 8; ++w) s += redsum[w];

  const float inv = 1.0f / s;
#pragma unroll
  for (int i = 0; i < 8; ++i) row[t + i * 256] = v[i] * inv;
}

// ---------------------------------------------------------------------------
// Driver: Q/K/V projections -> batched Q.K^T -> softmax -> attn.V -> out GEMM
// Workspace layout (floats): Q(8M) | K(8M) | V(8M) | scores(16M)  = 160 MB.
// ctx reuses the Q buffer (Q is dead after the score GEMM).
// ---------------------------------------------------------------------------
extern "C" void kernel_launch(void* const* d_in, const int* in_sizes, int n_in,
                              void* d_out, int out_size, void* d_ws, size_t ws_size,
                              hipStream_t stream) {
  (void)in_sizes; (void)n_in; (void)out_size; (void)ws_size;

  const float* query = (const float*)d_in[0];
  const float* keyval = (const float*)d_in[1];
  const float* Wq = (const float*)d_in[2];
  const float* bq = (const float*)d_in[3];
  const float* Wk = (const float*)d_in[4];
  const float* bk = (const float*)d_in[5];
  const float* Wv = (const float*)d_in[6];
  const float* bv = (const float*)d_in[7];
  const float* Wo = (const float*)d_in[8];
  const float* bo = (const float*)d_in[9];
  float* out = (float*)d_out;

  const int Bz = 4, S = 2048, D = 1024;
  const long long SD  = (long long)S * D;     // 2M floats
  const long long MSD = (long long)Bz * SD;   // 8M floats
  const long long SS  = (long long)S * S;     // 4M floats per batch

  float* ws = (float*)d_ws;
  float* Qw = ws;
  float* Kw = ws + MSD;
  float* Vw = ws + 2 * MSD;
  float* Sw = ws + 3 * MSD;  // Bz * SS floats
  float* Cw = Qw;            // ctx reuses Q

  const dim3 blk(256);
  const int Mflat = Bz * S;  // 8192

  // --- projections: [8192 x 1024] = in @ W + b ---
  const dim3 gproj(Mflat / BM, D / BN, 1);
  wmma_gemm_f32<false, true><<<gproj, blk, 0, stream>>>(
      query, Wq, bq, Qw, D, D, D, D, 0LL, 0LL, 0LL);
  wmma_gemm_f32<false, true><<<gproj, blk, 0, stream>>>(
      keyval, Wk, bk, Kw, D, D, D, D, 0LL, 0LL, 0LL);
  wmma_gemm_f32<false, true><<<gproj, blk, 0, stream>>>(
      keyval, Wv, bv, Vw, D, D, D, D, 0LL, 0LL, 0LL);

  // --- scores[b] = Q[b] @ K[b]^T  (K row-major == B column-major) ---
  const dim3 gsc(S / BM, S / BN, Bz);
  wmma_gemm_f32<true, false><<<gsc, blk, 0, stream>>>(
      Qw, Kw, nullptr, Sw, D, D, D, S, SD, SD, SS);

  // --- softmax over last dim ---
  softmax_rows2048<<<Bz * S, blk, 0, stream>>>(Sw);

  // --- ctx[b] = attn[b] @ V[b] ---
  const dim3 gctx(S / BM, D / BN, Bz);
  wmma_gemm_f32<false, false><<<gctx, blk, 0, stream>>>(
      Sw, Vw, nullptr, Cw, S, S, D, D, SS, SD, SD);

  // --- out = ctx @ Wo + bo ---
  wmma_gemm_f32<false, true><<<gproj, blk, 0, stream>>>(
      Cw, Wo, bo, out, D, D, D, D, 0LL, 0LL, 0LL);
}